// GatedGCN_8409545966055
// MI455X (gfx1250) — compile-verified
//
#include <hip/hip_runtime.h>
#include <math.h>

#define Dm 128   // latent dim (fixed by reference)

typedef float v2f __attribute__((ext_vector_type(2)));
typedef float v8f __attribute__((ext_vector_type(8)));

// ---------------------------------------------------------------------------
// GEMM: Y[M x 128] = X[M x 128] @ W^T + bias      (W is [128 x 128] row-major)
// out[m,i] = sum_k X[m,k] * W[i,k] + bias[i]
// One wave per 16-row tile, 8 column tiles (16 wide) per wave.
// Uses V_WMMA_F32_16X16X4_F32 (full fp32, matches reference precision).
// f32 WMMA layouts (ISA 7.12.2):
//   A 16x4 : lane L -> row M=L%16 ; VGPR0/1 = K = 2*(L/16) + {0,1}
//   B 4x16 : lane L -> col N=L%16 ; VGPR0/1 = K = 2*(L/16) + {0,1}
//   C/D    : lane L, VGPR v -> row M = v + 8*(L/16), col N = L%16
// ---------------------------------------------------------------------------
__global__ __launch_bounds__(128) void ggcn_gemm_xWt_bias(
    const float* __restrict__ X, const float* __restrict__ W,
    const float* __restrict__ bias, float* __restrict__ Y, int M)
{
    __shared__ __align__(16) float tile[64 * Dm];   // 32 KB: 64 rows of X
    const int tid  = threadIdx.x;
    const int wave = tid >> 5;
    const int lane = tid & 31;
    const int mblk = blockIdx.x * 64;

    // Cooperative, coalesced load of 64 rows (guard tail rows with zero-fill).
    const float4* Xv = (const float4*)X;            // 32 float4 per row
    float4* Tv = (float4*)tile;
    #pragma unroll
    for (int i = 0; i < 16; ++i) {
        int idx = i * 128 + tid;                    // 0..2047 float4 slots
        int row = mblk + (idx >> 5);
        float4 v;
        if (row < M) v = Xv[(size_t)row * 32 + (idx & 31)];
        else { v.x = 0.f; v.y = 0.f; v.z = 0.f; v.w = 0.f; }
        Tv[idx] = v;
    }
    __syncthreads();

    const int m0g = mblk + wave * 16;               // global first row of tile
    if (m0g < M) {                                  // wave-uniform: EXEC all-1s inside
        const int hf = lane >> 4;                   // lane half (K split)
        const int lr = lane & 15;

        v8f acc[8];
        #pragma unroll
        for (int t = 0; t < 8; ++t) {
            float bv = bias[t * 16 + lr];           // bias depends on column only
            #pragma unroll
            for (int v = 0; v < 8; ++v) acc[t][v] = bv;
        }

        const float* arow = &tile[(wave * 16 + lr) * Dm];
        #pragma unroll 2
        for (int k0 = 0; k0 < Dm; k0 += 4) {
            v2f a;
            a.x = arow[k0 + 2 * hf + 0];
            a.y = arow[k0 + 2 * hf + 1];
            #pragma unroll
            for (int t = 0; t < 8; ++t) {
                const float* wr = W + (size_t)(t * 16 + lr) * Dm + k0 + 2 * hf;
                v2f b; b.x = wr[0]; b.y = wr[1];
                acc[t] = __builtin_amdgcn_wmma_f32_16x16x4_f32(
                    false, a, false, b, (short)0, acc[t], false, false);
            }
        }

        float* yb = Y + (size_t)m0g * Dm;
        #pragma unroll
        for (int t = 0; t < 8; ++t)
            #pragma unroll
            for (int v = 0; v < 8; ++v)
                yb[(size_t)(v + 8 * hf) * Dm + t * 16 + lr] = acc[t][v];
    }
}

// ---------------------------------------------------------------------------
// Pass 1 over edges: build ne_f = A[s0]+B[s1]+C and ne_b = A[s1]+B[s0]+C,
// writing ne_b in-place over C; accumulate per-channel BN sums/sumsq for both.
// stat layout: [sum_f(128), sq_f(128), sum_b(128), sq_b(128)]
// ---------------------------------------------------------------------------
__global__ __launch_bounds__(256) void ggcn_edge_ne_stats(
    const int* __restrict__ s0, const int* __restrict__ s1,
    const float* __restrict__ Abuf, const float* __restrict__ Bbuf,
    float* __restrict__ Cneb,      // in: C   out: ne_b (in-place)
    float* __restrict__ nef,       // out: ne_f
    float* __restrict__ stat, int E)
{
    const int c   = threadIdx.x & (Dm - 1);
    const int sub = threadIdx.x >> 7;              // 0 or 1
    const int r0  = blockIdx.x * 64;
    float sf = 0.f, qf = 0.f, sb = 0.f, qb = 0.f;
    for (int i = sub; i < 64; i += 2) {
        int r = r0 + i;
        if (r >= E) break;
        int a = s0[r], d = s1[r];
        size_t o = (size_t)r * Dm + c;
        float Cv = Cneb[o];
        float f  = Abuf[(size_t)a * Dm + c] + Bbuf[(size_t)d * Dm + c] + Cv;
        float bk = Abuf[(size_t)d * Dm + c] + Bbuf[(size_t)a * Dm + c] + Cv;
        nef[o]  = f;
        Cneb[o] = bk;
        sf += f;  qf += f * f;
        sb += bk; qb += bk * bk;
    }
    __shared__ float red[4][256];
    red[0][threadIdx.x] = sf; red[1][threadIdx.x] = qf;
    red[2][threadIdx.x] = sb; red[3][threadIdx.x] = qb;
    __syncthreads();
    if (sub == 0) {
        atomicAdd(&stat[0 * Dm + c], sf + red[0][threadIdx.x + 128]);
        atomicAdd(&stat[1 * Dm + c], qf + red[1][threadIdx.x + 128]);
        atomicAdd(&stat[2 * Dm + c], sb + red[2][threadIdx.x + 128]);
        atomicAdd(&stat[3 * Dm + c], qb + red[3][threadIdx.x + 128]);
    }
}

// stat: pairs of (sum, sumsq) -> (mean, rsqrt(var+eps)); biased variance.
__global__ void ggcn_bn_finalize(float* stat, float inv_n, int pairs)
{
    int c = threadIdx.x;                            // 128 threads
    for (int p = 0; p < pairs; ++p) {
        float mean = stat[(2 * p + 0) * Dm + c] * inv_n;
        float var  = stat[(2 * p + 1) * Dm + c] * inv_n - mean * mean;
        stat[(2 * p + 0) * Dm + c] = mean;
        stat[(2 * p + 1) * Dm + c] = rsqrtf(var + 1e-5f);
    }
}

// ---------------------------------------------------------------------------
// Pass 2 over edges: apply BN+ReLU+residual to both gates, store e_new
// in-place over ne_f, fuse the two gate messages into ONE atomic scatter:
//   agg[s1] += Vf[s0]*sf/(sf+1e-6) + Vb[s0]*sb/(sb+1e-6)
// (agg was pre-initialized with U by the GEMM.)
// ---------------------------------------------------------------------------
__global__ __launch_bounds__(256) void ggcn_edge_apply_scatter(
    const int* __restrict__ s0, const int* __restrict__ s1,
    float* __restrict__ nef,            // in: ne_f  out: e_new (in-place)
    const float* __restrict__ neb, const float* __restrict__ ecur,
    const float* __restrict__ Vf, const float* __restrict__ Vb,
    const float* __restrict__ g, const float* __restrict__ bt,
    const float* __restrict__ stat,     // mean_f, rstd_f, mean_b, rstd_b
    float* __restrict__ agg, int E)
{
    const int c   = threadIdx.x & (Dm - 1);
    const int sub = threadIdx.x >> 7;
    const float mf = stat[c],          rf = stat[Dm + c];
    const float mb = stat[2 * Dm + c], rb = stat[3 * Dm + c];
    const float gg = g[c], bb = bt[c];
    const int r0 = blockIdx.x * 64;
    for (int i = sub; i < 64; i += 2) {
        int r = r0 + i;
        if (r >= E) break;
        size_t o = (size_t)r * Dm + c;
        int a = s0[r], d = s1[r];
        float ev = ecur[o];
        float nf = fmaxf((nef[o] - mf) * rf * gg + bb, 0.f) + ev;
        float nb = fmaxf((neb[o] - mb) * rb * gg + bb, 0.f) + ev;
        nef[o] = nf;                    // becomes next-layer e
        float sf = 1.f / (1.f + expf(-nf));
        float sb = 1.f / (1.f + expf(-nb));
        float msg = Vf[(size_t)a * Dm + c] * (sf / (sf + 1e-6f))
                  + Vb[(size_t)a * Dm + c] * (sb / (sb + 1e-6f));
        atomicAdd(&agg[(size_t)d * Dm + c], msg);
    }
}

// Per-channel stats over node tensor (for h BatchNorm).
__global__ __launch_bounds__(256) void ggcn_row_stats(
    const float* __restrict__ X, float* __restrict__ stat, int M)
{
    const int c = threadIdx.x & (Dm - 1);
    const int sub = threadIdx.x >> 7;
    const int r0 = blockIdx.x * 64;
    float s = 0.f, q = 0.f;
    for (int i = sub; i < 64; i += 2) {
        int r = r0 + i;
        if (r >= M) break;
        float v = X[(size_t)r * Dm + c];
        s += v; q += v * v;
    }
    __shared__ float red[2][256];
    red[0][threadIdx.x] = s; red[1][threadIdx.x] = q;
    __syncthreads();
    if (sub == 0) {
        atomicAdd(&stat[c],      s + red[0][threadIdx.x + 128]);
        atomicAdd(&stat[Dm + c], q + red[1][threadIdx.x + 128]);
    }
}

// h_new = relu(BN(agg)) + h_cur
__global__ void ggcn_node_apply(
    const float* __restrict__ agg, const float* __restrict__ hcur,
    const float* __restrict__ g, const float* __restrict__ bt,
    const float* __restrict__ stat, float* __restrict__ hout, size_t n)
{
    size_t i = (size_t)blockIdx.x * blockDim.x + threadIdx.x;
    if (i >= n) return;
    int c = (int)(i & (Dm - 1));
    float v = (agg[i] - stat[c]) * stat[Dm + c] * g[c] + bt[c];
    hout[i] = fmaxf(v, 0.f) + hcur[i];
}

__global__ void ggcn_zero(float* p, int n)
{
    int i = blockIdx.x * blockDim.x + threadIdx.x;
    if (i < n) p[i] = 0.f;
}

// ---------------------------------------------------------------------------
extern "C" void kernel_launch(void* const* d_in, const int* in_sizes, int n_in,
                              void* d_out, int out_size, void* d_ws, size_t ws_size,
                              hipStream_t stream)
{
    const int N  = in_sizes[1] / Dm;            // nodes
    const int E  = in_sizes[2] / Dm;            // edges
    const int L  = in_sizes[4] / Dm;            // layers (bA is [L,D])
    const int DD = Dm * Dm;

    const int*   ei   = (const int*)  d_in[0];
    const int*   s0   = ei;
    const int*   s1   = ei + E;
    const float* h_in = (const float*)d_in[1];
    const float* e_in = (const float*)d_in[2];
    const float* WA = (const float*)d_in[3],  *bA = (const float*)d_in[4];
    const float* WB = (const float*)d_in[5],  *bB = (const float*)d_in[6];
    const float* WC = (const float*)d_in[7],  *bC = (const float*)d_in[8];
    const float* WU = (const float*)d_in[9],  *bU = (const float*)d_in[10];
    const float* WVf= (const float*)d_in[11], *bVf= (const float*)d_in[12];
    const float* WVb= (const float*)d_in[13], *bVb= (const float*)d_in[14];
    const float* bnh_g = (const float*)d_in[15], *bnh_b = (const float*)d_in[16];
    const float* bne_g = (const float*)d_in[17], *bne_b = (const float*)d_in[18];

    const size_t ND = (size_t)N * Dm, ED = (size_t)E * Dm;
    float* ws   = (float*)d_ws;
    float* H0   = ws;            float* H1   = H0 + ND;
    float* Abuf = H1 + ND;       float* Bbuf = Abuf + ND;
    float* Vf   = Bbuf + ND;     float* Vb   = Vf + ND;
    float* agg  = Vb + ND;
    float* E0   = agg + ND;      float* E1   = E0 + ED;
    float* E2   = E1 + ED;                      // C / ne_b scratch
    float* stat = E2 + ED;                      // 4*Dm floats
    float* h_out = (float*)d_out;
    float* e_out = h_out + ND;

    const int gN = (N + 63) / 64;               // GEMM / stats blocks (nodes)
    const int gE = (E + 63) / 64;               // blocks over edges
    const int gNA = (int)((ND + 255) / 256);

    const float* hcur = h_in;
    const float* ecur = e_in;

    for (int l = 0; l < L; ++l) {
        const size_t wo = (size_t)l * DD, bo = (size_t)l * Dm;
        float* hnext = (l == L - 1) ? h_out : ((l & 1) ? H1 : H0);
        float* nef   = (l == L - 1) ? e_out : ((l & 1) ? E1 : E0);

        // Dense projections (f32 WMMA). U lands directly in agg.
        ggcn_gemm_xWt_bias<<<gN, 128, 0, stream>>>(hcur, WA + wo, bA + bo, Abuf, N);
        ggcn_gemm_xWt_bias<<<gN, 128, 0, stream>>>(hcur, WB + wo, bB + bo, Bbuf, N);
        ggcn_gemm_xWt_bias<<<gN, 128, 0, stream>>>(hcur, WU + wo, bU + bo, agg,  N);
        ggcn_gemm_xWt_bias<<<gN, 128, 0, stream>>>(hcur, WVf+ wo, bVf+ bo, Vf,   N);
        ggcn_gemm_xWt_bias<<<gN, 128, 0, stream>>>(hcur, WVb+ wo, bVb+ bo, Vb,   N);
        ggcn_gemm_xWt_bias<<<gE, 128, 0, stream>>>(ecur, WC + wo, bC + bo, E2,   E);

        // Edge pass 1: ne_f/ne_b + BN stats (both gates).
        ggcn_zero<<<2, 256, 0, stream>>>(stat, 4 * Dm);
        ggcn_edge_ne_stats<<<gE, 256, 0, stream>>>(s0, s1, Abuf, Bbuf, E2, nef, stat, E);
        ggcn_bn_finalize<<<1, Dm, 0, stream>>>(stat, 1.0f / (float)E, 2);

        // Edge pass 2: BN+ReLU+residual, e_new in-place, fused dual-gate scatter.
        ggcn_edge_apply_scatter<<<gE, 256, 0, stream>>>(
            s0, s1, nef, E2, ecur, Vf, Vb, bne_g + bo, bne_b + bo, stat, agg, E);

        // Node BN + residual.
        ggcn_zero<<<1, 256, 0, stream>>>(stat, 2 * Dm);
        ggcn_row_stats<<<gN, 256, 0, stream>>>(agg, stat, N);
        ggcn_bn_finalize<<<1, Dm, 0, stream>>>(stat, 1.0f / (float)N, 1);
        ggcn_node_apply<<<gNA, 256, 0, stream>>>(
            agg, hcur, bnh_g + bo, bnh_b + bo, stat, hnext, ND);

        hcur = hnext;
        ecur = nef;
    }
}